// AEDecoder_66340064854755
// MI455X (gfx1250) — compile-verified
//
#include <hip/hip_runtime.h>

// ---------------- problem constants (from reference) ----------------
#define N_TF     1024
#define N_GENES  20000
#define W        4
#define FANIN    16
#define B_BATCH  128
#define M_MID    (N_GENES * W)          // 80000

// ---------------- tiling ----------------
#define GENES_PER_BLOCK 16
#define NODES_PER_BLOCK (GENES_PER_BLOCK * W)   // 64 mid-nodes per block
#define H1_PAD 136     // padded h1 row (floats): 2*H1_PAD % 64 == 16 -> disjoint bank
                       // groups for the two half-wave B-operand reads
#define OPAD   17      // padded out-tile row (floats): stride 17 -> conflict-free
#define W2_LDS (GENES_PER_BLOCK * W * W)        // 256 staged layer-2 weights

typedef float v2f __attribute__((ext_vector_type(2)));
typedef float v8f __attribute__((ext_vector_type(8)));

// Branch-free tanh on the HW transcendentals:
// tanh(x) = sign(x) * (1 - 2/(exp2(2*log2e*|x|) + 1))
// |x| large -> exp2 = +inf -> rcp = 0 -> result = +/-1 (correct saturation).
__device__ __forceinline__ float fast_tanhf(float x) {
  const float ax = __builtin_fabsf(x);
  const float e  = __builtin_amdgcn_exp2f(ax * 2.8853900817779268f); // 2/ln2
  const float r  = 1.0f - 2.0f * __builtin_amdgcn_rcpf(e + 1.0f);
  return __builtin_copysignf(r, x);
}

// features [B, N_TF] -> xT [N_TF, B]  (512 KB, stays L2-resident; makes every
// edge gather a contiguous 512B row, coalesced across batch lanes)
__global__ void tf_transpose_kernel(const float* __restrict__ features,
                                    float* __restrict__ xT) {
  int g = blockIdx.x * blockDim.x + threadIdx.x;   // N_TF*B threads
  int t = g >> 7;          // TF index
  int b = g & 127;         // batch index
  xT[t * B_BATCH + b] = features[b * N_TF + t];
}

__launch_bounds__(256)
__global__ void gene_mlp_fused_kernel(const float* __restrict__ xT,
                                      const float* __restrict__ w1,
                                      const float* __restrict__ b1,
                                      const int*   __restrict__ in1,
                                      const float* __restrict__ w2,
                                      const float* __restrict__ b2,
                                      const float* __restrict__ w3,
                                      const float* __restrict__ b3,
                                      float* __restrict__ out) {
  __shared__ float h1[NODES_PER_BLOCK * H1_PAD];   // [node_local][batch] layer-1 act
  __shared__ float ot[B_BATCH * OPAD];             // [batch][gene_local] outputs
  __shared__ float w2s[W2_LDS];                    // staged layer-2 4x4 blocks

  const int tid   = threadIdx.x;
  const int wave  = tid >> 5;        // 0..7
  const int lane  = tid & 31;
  const int gene0 = blockIdx.x * GENES_PER_BLOCK;  // first gene of this block
  const int node0 = gene0 * W;                     // first mid-node

  // stage this block's w2 (16 genes x 4x4 = 256 floats), reused by all waves
  w2s[tid] = w2[node0 * W + tid];

  // ---------------- Phase A: sparse layer 1 (gather + FMA + tanh) ----------
  // wave-uniform node => in1/w1/b1 become scalar (s_load) traffic;
  // each lane owns 4 consecutive batch elements -> one b128 gather per edge.
  for (int it = 0; it < NODES_PER_BLOCK / 8; ++it) {
    const int nl = wave + 8 * it;        // local node 0..63
    const int n  = node0 + nl;           // global mid-node
    const int eb = n * FANIN;
    const float bias = b1[n];
    float4 acc = { bias, bias, bias, bias };
    for (int k = 0; k < FANIN; ++k) {
      const int   idx = in1[eb + k];     // wave-uniform
      const float wv  = w1[eb + k];      // wave-uniform
      const float4 xv = *(const float4*)(xT + idx * B_BATCH + 4 * lane);
      if (k + 1 < FANIN) {               // hide random-gather latency
        const int idx2 = in1[eb + k + 1];
        __builtin_prefetch(xT + idx2 * B_BATCH + 4 * lane, 0, 0);
      }
      acc.x += wv * xv.x; acc.y += wv * xv.y;
      acc.z += wv * xv.z; acc.w += wv * xv.w;
    }
    float4 h;
    h.x = fast_tanhf(acc.x); h.y = fast_tanhf(acc.y);
    h.z = fast_tanhf(acc.z); h.w = fast_tanhf(acc.w);
    *(float4*)(&h1[nl * H1_PAD + 4 * lane]) = h;
  }
  __syncthreads();

  // ---------------- Phase B: layer 2 via V_WMMA_F32_16X16X4_F32 ------------
  // D[m,n] = sum_k w2[(4g+m)*4+k] * h1[4g+k][btile*16+n]   (K=4 matches shape)
  // A 16x4 layout: lanes 0-15 hold K=0,1 ; lanes 16-31 hold K=2,3
  // C/D layout:    VGPR r -> row M=r (lanes 0-15); rows 4..15 of A are zero.
  {
    const int m    = lane & 15;          // matrix row (A) / column n (B,D)
    const int half = lane >> 4;
    const int k0   = 2 * half;           // K pair held by this half-wave
    const int bt   = wave;               // batch tile 0..7 (16 batch each)
    const int mc   = (m < W) ? m : (W - 1);   // clamped row for safe loads

    for (int gl = 0; gl < GENES_PER_BLOCK; ++gl) {
      const int gg = gene0 + gl;         // global gene

      // B operand: h1 rows (in-nodes k0,k0+1) x 16 batch columns (2addr LDS)
      const float* hrow = &h1[(gl * W + k0) * H1_PAD + bt * 16 + m];
      v2f bv; bv.x = hrow[0]; bv.y = hrow[H1_PAD];

      // A operand: w2 row m (zero rows m>=4). Unconditional aligned b64 LDS
      // load from a clamped address + v_cndmask -> no EXEC manipulation.
      const float2 wt = *(const float2*)&w2s[(gl * W + mc) * W + k0];
      v2f av;
      av.x = (m < W) ? wt.x : 0.0f;
      av.y = (m < W) ? wt.y : 0.0f;

      v8f c = {};
      c = __builtin_amdgcn_wmma_f32_16x16x4_f32(
              /*neg_a=*/false, av, /*neg_b=*/false, bv,
              /*c_mod=*/(short)0, c, /*reuse_a=*/false, /*reuse_b=*/false);

      // fused layer 3: out[b,g] = b3 + sum_r w3[4g+r]*tanh(D[r,b] + b2[4g+r])
      float accn = b3[gg];
#pragma unroll
      for (int r = 0; r < W; ++r) {
        const float h2 = fast_tanhf(c[r] + b2[gg * W + r]);  // scalar b2/w3
        accn += w3[gg * W + r] * h2;
      }
      if (half == 0)                     // lanes 0-15 hold valid rows 0..3
        ot[(bt * 16 + m) * OPAD + gl] = accn;
    }
  }
  __syncthreads();

  // ---------------- Phase C: coalesced output stores -----------------------
  // lanes sweep 16 contiguous genes of one output row (64B per half-wave).
  {
    const int g  = tid & 15;
    const int br = tid >> 4;             // 0..15
    for (int it = 0; it < 8; ++it) {
      const int b = br + 16 * it;        // 0..127
      out[b * N_GENES + gene0 + g] = ot[b * OPAD + g];
    }
  }
}

extern "C" void kernel_launch(void* const* d_in, const int* in_sizes, int n_in,
                              void* d_out, int out_size, void* d_ws, size_t ws_size,
                              hipStream_t stream) {
  const float* features = (const float*)d_in[0];   // [B, N_TF]
  const float* w1       = (const float*)d_in[1];   // [M*FANIN]
  const float* b1       = (const float*)d_in[2];   // [M]
  const float* w2       = (const float*)d_in[3];   // [M*W]
  const float* b2       = (const float*)d_in[4];   // [M]
  const float* w3       = (const float*)d_in[5];   // [N_GENES*W]
  const float* b3       = (const float*)d_in[6];   // [N_GENES]
  // d_in[7] = out1 (implicit), d_in[8] = in1, d_in[9..12] structured/unused
  const int*   in1      = (const int*)d_in[8];

  float* out = (float*)d_out;                      // [B, N_GENES]
  float* xT  = (float*)d_ws;                       // [N_TF, B] = 512 KB scratch

  // transpose features so edge gathers are contiguous across the batch
  tf_transpose_kernel<<<(N_TF * B_BATCH) / 256, 256, 0, stream>>>(features, xT);

  // fused 3-layer sparse MLP, 16 genes per block
  gene_mlp_fused_kernel<<<N_GENES / GENES_PER_BLOCK, 256, 0, stream>>>(
      xT, w1, b1, in1, w2, b2, w3, b3, out);
}